// Bpnet_81449759801462
// MI455X (gfx1250) — compile-verified
//
#include <hip/hip_runtime.h>

typedef __attribute__((ext_vector_type(2))) float v2f;
typedef __attribute__((ext_vector_type(8))) float v8f;

#define BP_EPS 1e-8f

// ---------------------------------------------------------------------------
// Kernel A: per 16-edge tile (one wave32 per tile):
//   temp2[tile] = log(cav[tile] @ C + eps)   via 4x V_WMMA_F32_16X16X4_F32
//   marg[edge_dst[e]] += temp2[e]            via global_atomic_add_f32
// A operand layout (f32 16x4): lanes 0-15 hold K=kk..kk+1 of row M=lane,
// lanes 16-31 hold K=kk+2..kk+3 (kk = 2*(lane>>4)). B (=C matrix) symmetric.
// D layout: VGPR v of lane L = element (row v + 8*(L>>4), col L&15).
// All flat indices fit in int (E*16 = 25.6M) -> 32-bit saddr+voffset forms.
// ---------------------------------------------------------------------------
__global__ __launch_bounds__(256)
void bp_edge_gemm_scatter(const float* __restrict__ cav,   // [E,16]
                          const float* __restrict__ Cm,    // [16,16] row-major
                          const int*   __restrict__ edge_dst,
                          float* __restrict__ temp2,       // [E,16]
                          float* __restrict__ marg,        // [N,16] atomic target
                          int E, int ntiles) {
    int gwave = (int)((blockIdx.x * blockDim.x + threadIdx.x) >> 5);
    if (gwave >= ntiles) return;                 // wave-uniform branch
    int lane = threadIdx.x & 31;
    int n    = lane & 15;                        // column / A-row index
    int hi   = lane >> 4;                        // 0 or 1
    int kk   = hi * 2;                           // K sub-offset in each K4 chunk

    int  base = gwave << 4;
    bool full = (base + 16) <= E;                // wave-uniform
    int  erow = base + n;

    v2f a0 = {0.f, 0.f}, a1 = a0, a2 = a0, a3 = a0;
    if (full || erow < E) {
        const float* ap = cav + (size_t)erow * 16 + kk;
        a0 = *(const v2f*)(ap + 0);
        a1 = *(const v2f*)(ap + 4);
        a2 = *(const v2f*)(ap + 8);
        a3 = *(const v2f*)(ap + 12);
    }

    v2f b0, b1, b2, b3;
    b0.x = Cm[(kk + 0)  * 16 + n];  b0.y = Cm[(kk + 1)  * 16 + n];
    b1.x = Cm[(kk + 4)  * 16 + n];  b1.y = Cm[(kk + 5)  * 16 + n];
    b2.x = Cm[(kk + 8)  * 16 + n];  b2.y = Cm[(kk + 9)  * 16 + n];
    b3.x = Cm[(kk + 12) * 16 + n];  b3.y = Cm[(kk + 13) * 16 + n];

    v8f acc = {0.f, 0.f, 0.f, 0.f, 0.f, 0.f, 0.f, 0.f};
    acc = __builtin_amdgcn_wmma_f32_16x16x4_f32(false, a0, false, b0, (short)0, acc, false, false);
    acc = __builtin_amdgcn_wmma_f32_16x16x4_f32(false, a1, false, b1, (short)0, acc, false, false);
    acc = __builtin_amdgcn_wmma_f32_16x16x4_f32(false, a2, false, b2, (short)0, acc, false, false);
    acc = __builtin_amdgcn_wmma_f32_16x16x4_f32(false, a3, false, b3, (short)0, acc, false, false);

    int mrow0 = hi * 8;
    if (full) {
        // fast path: no per-element bounds branches
        #pragma unroll
        for (int v = 0; v < 8; ++v) {
            int e = base + mrow0 + v;
            float val = __logf(acc[v] + BP_EPS);
            temp2[e * 16 + n] = val;
            int dst = edge_dst[e];
            __hip_atomic_fetch_add(&marg[dst * 16 + n], val,
                                   __ATOMIC_RELAXED, __HIP_MEMORY_SCOPE_AGENT);
        }
    } else {
        #pragma unroll
        for (int v = 0; v < 8; ++v) {
            int e = base + mrow0 + v;
            if (e < E) {
                float val = __logf(acc[v] + BP_EPS);
                temp2[e * 16 + n] = val;
                int dst = edge_dst[e];
                __hip_atomic_fetch_add(&marg[dst * 16 + n], val,
                                       __ATOMIC_RELAXED, __HIP_MEMORY_SCOPE_AGENT);
            }
        }
    }
}

// ---------------------------------------------------------------------------
// Kernel B: cav[e] = softmax(marg[a1_src[e]] - temp2[indice_ij[e]])
// one thread per edge; 16B vector gathers (rows are 64B aligned)
// ---------------------------------------------------------------------------
__global__ __launch_bounds__(256)
void bp_edge_softmax(const float* __restrict__ marg,    // [N,16]
                     const float* __restrict__ temp2,   // [E,16]
                     const int*   __restrict__ a1_src,
                     const int*   __restrict__ indice_ij,
                     float* __restrict__ cav_out,       // [E,16]
                     int E) {
    int e = blockIdx.x * blockDim.x + threadIdx.x;
    if (e >= E) return;
    const float4* mrow = (const float4*)(marg  + (size_t)a1_src[e]    * 16);
    const float4* trow = (const float4*)(temp2 + (size_t)indice_ij[e] * 16);
    float x[16];
    #pragma unroll
    for (int j = 0; j < 4; ++j) {
        float4 m = mrow[j], t = trow[j];
        x[4*j+0] = m.x - t.x;  x[4*j+1] = m.y - t.y;
        x[4*j+2] = m.z - t.z;  x[4*j+3] = m.w - t.w;
    }
    float mx = x[0];
    #pragma unroll
    for (int i = 1; i < 16; ++i) mx = fmaxf(mx, x[i]);
    float s = 0.f;
    #pragma unroll
    for (int i = 0; i < 16; ++i) { x[i] = __expf(x[i] - mx); s += x[i]; }
    float inv = 1.0f / s;
    float4* orow = (float4*)(cav_out + (size_t)e * 16);
    #pragma unroll
    for (int j = 0; j < 4; ++j) {
        float4 o;
        o.x = x[4*j+0] * inv;  o.y = x[4*j+1] * inv;
        o.z = x[4*j+2] * inv;  o.w = x[4*j+3] * inv;
        orow[j] = o;
    }
}

// ---------------------------------------------------------------------------
// Final: out = log_softmax(marg, axis=1); one thread per node
// ---------------------------------------------------------------------------
__global__ __launch_bounds__(256)
void bp_node_logsoftmax(const float* __restrict__ marg, float* __restrict__ out, int N) {
    int i = blockIdx.x * blockDim.x + threadIdx.x;
    if (i >= N) return;
    const float4* mrow = (const float4*)(marg + (size_t)i * 16);
    float x[16];
    #pragma unroll
    for (int j = 0; j < 4; ++j) {
        float4 m = mrow[j];
        x[4*j+0] = m.x; x[4*j+1] = m.y; x[4*j+2] = m.z; x[4*j+3] = m.w;
    }
    float mx = x[0];
    #pragma unroll
    for (int k = 1; k < 16; ++k) mx = fmaxf(mx, x[k]);
    float s = 0.f;
    #pragma unroll
    for (int k = 0; k < 16; ++k) s += __expf(x[k] - mx);
    float lse = mx + __logf(s);
    float4* orow = (float4*)(out + (size_t)i * 16);
    #pragma unroll
    for (int j = 0; j < 4; ++j) {
        float4 o;
        o.x = x[4*j+0] - lse;  o.y = x[4*j+1] - lse;
        o.z = x[4*j+2] - lse;  o.w = x[4*j+3] - lse;
        orow[j] = o;
    }
}

extern "C" void kernel_launch(void* const* d_in, const int* in_sizes, int n_in,
                              void* d_out, int out_size, void* d_ws, size_t ws_size,
                              hipStream_t stream) {
    // inputs: marg_i[N,16] (dead), cav_ij[E,16], C[16,16], field_i[N,16],
    //         edge_dst[E], a1_src[E], indice_ij[E]
    const float* cav_in    = (const float*)d_in[1];
    const float* Cm        = (const float*)d_in[2];
    const float* field     = (const float*)d_in[3];
    const int*   edge_dst  = (const int*)d_in[4];
    const int*   a1_src    = (const int*)d_in[5];
    const int*   indice_ij = (const int*)d_in[6];
    float*       out       = (float*)d_out;

    const int N = in_sizes[0] / 16;
    const int E = in_sizes[1] / 16;

    // workspace partition: temp2 [E*16] | cav [E*16] | marg [N*16]
    float* wsf   = (float*)d_ws;
    float* temp2 = wsf;
    float* cavw  = wsf + (size_t)E * 16;
    float* marg  = wsf + (size_t)2 * E * 16;

    const int DEPTH   = 5;
    const int ntiles  = (E + 15) / 16;                    // 16 edges per wave32
    const int blkA    = 256;                              // 8 waves / block
    const int grdA    = (ntiles * 32 + blkA - 1) / blkA;
    const int grdB    = (E + 255) / 256;
    const int grdN    = (N + 255) / 256;
    const size_t margBytes = (size_t)N * 16 * sizeof(float);

    for (int d = 0; d < DEPTH; ++d) {
        // marg starts each step as field_i (segment_sum base)
        hipMemcpyAsync(marg, field, margBytes, hipMemcpyDeviceToDevice, stream);
        const float* cav_src = (d == 0) ? cav_in : cavw;
        bp_edge_gemm_scatter<<<grdA, blkA, 0, stream>>>(cav_src, Cm, edge_dst,
                                                        temp2, marg, E, ntiles);
        if (d < DEPTH - 1) {
            bp_edge_softmax<<<grdB, 256, 0, stream>>>(marg, temp2, a1_src,
                                                      indice_ij, cavw, E);
        }
    }
    bp_node_logsoftmax<<<grdN, 256, 0, stream>>>(marg, out, N);
}